// RegularizationLoss_34162169872832
// MI455X (gfx1250) — compile-verified
//
#include <hip/hip_runtime.h>
#include <hip/hip_bf16.h>
#include <math.h>
#include <stdint.h>

// ---------------------------------------------------------------------------
// RegularizationLoss on MI455X (gfx1250, wave32).
// KNN pairwise dot-products on the matrix pipes via V_WMMA_F32_16X16X4_F32
// (coordinate dim 3 padded to K=4; -2 scale folded into the A fragment).
// Candidate tiles staged global->LDS with GLOBAL_LOAD_ASYNC_TO_LDS_B128
// (ASYNCcnt-tracked), top-16 selection uses a per-row min filter so most
// tiles skip the insert path entirely.
// ---------------------------------------------------------------------------

typedef __attribute__((ext_vector_type(2))) float v2f;
typedef __attribute__((ext_vector_type(8))) float v8f;
typedef int v4i_vs __attribute__((vector_size(16)));   // matches builtin param type

#define K_NN 16
#define FILTER_SCALE 2.0f
#define SHARPNESS_SIGMA 0.75f
#define EPS_REF 1e-17f

#if defined(__has_builtin)
#if __has_builtin(__builtin_amdgcn_global_load_async_to_lds_b128) && \
    __has_builtin(__builtin_amdgcn_s_wait_asynccnt)
#define HAVE_ASYNC_LDS 1
#endif
#endif
#ifndef HAVE_ASYNC_LDS
#define HAVE_ASYNC_LDS 0
#endif

#define GADDR_B128(p) ((__attribute__((address_space(1))) v4i_vs*)(uintptr_t)(p))
#define LADDR_B128(p) ((__attribute__((address_space(3))) v4i_vs*)(p))

__device__ __forceinline__ float eps_denom(float x) {
    return (x < EPS_REF) ? EPS_REF : x;   // jnp.where(x < EPS, EPS, x)
}

// ---------------- KNN via WMMA 16x16x4 f32 ----------------------------------
// One wave per block; block owns 16 query rows of one batch.
// D tile = d2q[M] + d2c[N] + wmma(-2*Q, C); per-row top-16 kept in LDS.
__global__ __launch_bounds__(32)
void knn_kernel(const float* __restrict__ pts,
                float* __restrict__ out_dist, int* __restrict__ out_idx, int P) {
    const int b    = blockIdx.y;
    const int q0   = blockIdx.x * 16;
    const int lane = threadIdx.x;        // 0..31
    const int half = lane >> 4;          // 0: K-rows 0,1 | 1: K-rows 2,3
    const int l15  = lane & 15;

    const float* ptsB = pts + (size_t)b * P * 3;

    __shared__ __align__(16) float qtile[48];       // 16 query points
    __shared__ __align__(16) float ctile[48];       // 16 candidate points
    __shared__ __align__(16) float distT[16 * 20];  // [row][col], stride 20 (16B-aligned rows)
    __shared__ float topd[16][K_NN];
    __shared__ int   topi[16][K_NN];

    // ---- stage query tile (192B, 16B aligned) into LDS ----
#if HAVE_ASYNC_LDS
    if (lane < 12)
        __builtin_amdgcn_global_load_async_to_lds_b128(
            GADDR_B128(ptsB + (size_t)q0 * 3 + lane * 4),
            LADDR_B128(&qtile[lane * 4]), 0, 0);
    __builtin_amdgcn_s_wait_asynccnt(0);
#else
    if (lane < 12)
        ((float4*)qtile)[lane] = ((const float4*)(ptsB + (size_t)q0 * 3))[lane];
#endif
    if (half == 0) {
#pragma unroll
        for (int k = 0; k < K_NN; ++k) { topd[l15][k] = __builtin_inff(); topi[l15][k] = 0; }
    }
    __syncthreads();

    // A fragment (16x4 f32, 2 VGPRs), scaled by -2 so WMMA yields -2*(q.c).
    // lanes 0-15 -> K=0,1 ; lanes 16-31 -> K=2,3 (K=3 padded with 0).
    v2f afrag;
    afrag.x = -2.0f * qtile[l15 * 3 + 2 * half];
    afrag.y = half ? 0.0f : (-2.0f * qtile[l15 * 3 + 1]);

    // d2 of query rows seen by this lane's C/D slots: slot j -> M = j + 8*half
    float d2q[8];
#pragma unroll
    for (int j = 0; j < 8; ++j) {
        const int m = j + 8 * half;
        const float x = qtile[m * 3 + 0], y = qtile[m * 3 + 1], z = qtile[m * 3 + 2];
        d2q[j] = x * x + y * y + z * z;
    }

    float worst = __builtin_inff();
    int   warg  = 0;

    for (int c0 = 0; c0 < P; c0 += 16) {
        // ---- stage candidate tile into LDS ----
#if HAVE_ASYNC_LDS
        if (lane < 12)
            __builtin_amdgcn_global_load_async_to_lds_b128(
                GADDR_B128(ptsB + (size_t)c0 * 3 + lane * 4),
                LADDR_B128(&ctile[lane * 4]), 0, 0);
        __builtin_amdgcn_s_wait_asynccnt(0);
#else
        if (lane < 12)
            ((float4*)ctile)[lane] = ((const float4*)(ptsB + (size_t)c0 * 3))[lane];
#endif
        if (c0 + 16 < P)   // pull next tile toward the WGP while we compute
            __builtin_prefetch(ptsB + (size_t)(c0 + 16 + l15) * 3, 0, 3);
        __syncthreads();

        // B fragment (4x16 f32): branchless build from LDS
        v2f bfrag;
        bfrag.x = ctile[l15 * 3 + 2 * half];
        bfrag.y = half ? 0.0f : ctile[l15 * 3 + 1];

        const float cx = ctile[l15 * 3 + 0], cy = ctile[l15 * 3 + 1], cz = ctile[l15 * 3 + 2];
        const float d2c = cx * cx + cy * cy + cz * cz;   // column N = lane&15

        v8f acc = {0.f, 0.f, 0.f, 0.f, 0.f, 0.f, 0.f, 0.f};
        acc = __builtin_amdgcn_wmma_f32_16x16x4_f32(
            /*neg_a=*/false, afrag, /*neg_b=*/false, bfrag,
            /*c_mod=*/(short)0, acc, /*reuse_a=*/false, /*reuse_b=*/false);

#pragma unroll
        for (int j = 0; j < 8; ++j)
            distT[(j + 8 * half) * 20 + l15] = d2q[j] + d2c + acc[j];
        __syncthreads();

        if (half == 0) {
            const int r  = l15;
            const int qg = q0 + r;
            // read the whole row as 4x b128, min-filter before any insert
            const float4* rowp = (const float4*)&distT[r * 20];
            const float4 v0 = rowp[0], v1 = rowp[1], v2 = rowp[2], v3 = rowp[3];
            float rv[16] = {v0.x, v0.y, v0.z, v0.w, v1.x, v1.y, v1.z, v1.w,
                            v2.x, v2.y, v2.z, v2.w, v3.x, v3.y, v3.z, v3.w};
            float m = rv[0];
#pragma unroll
            for (int cc = 1; cc < 16; ++cc) m = fminf(m, rv[cc]);
            m = fmaxf(m, 0.0f);

            if (m < worst) {                      // tile can improve the top-16
#pragma unroll 4
                for (int cc = 0; cc < 16; ++cc) {
                    const int ci = c0 + cc;
                    const float dv = fmaxf(rv[cc], 0.0f);   // clamp like reference
                    if (dv < worst && ci != qg) {           // exclude self
                        topd[r][warg] = dv;
                        topi[r][warg] = ci;
                        worst = topd[r][0]; warg = 0;
#pragma unroll
                        for (int k = 1; k < K_NN; ++k) {
                            const float t = topd[r][k];
                            if (t > worst) { worst = t; warg = k; }
                        }
                    }
                }
            }
        }
        __syncthreads();
    }

    // Ascending sort (slot 0 = nearest neighbor feeds s and the ball mask)
    if (half == 0) {
        const int r = l15;
        for (int i = 0; i < K_NN - 1; ++i) {
            int   mb = i;
            float md = topd[r][i];
            for (int k = i + 1; k < K_NN; ++k) {
                const float t = topd[r][k];
                if (t < md) { md = t; mb = k; }
            }
            if (mb != i) {
                const float td = topd[r][i]; topd[r][i] = md;          topd[r][mb] = td;
                const int   ti = topi[r][i]; topi[r][i] = topi[r][mb]; topi[r][mb] = ti;
            }
        }
        const size_t base = ((size_t)b * P + q0 + r) * K_NN;
#pragma unroll
        for (int k = 0; k < K_NN; ++k) {
            out_dist[base + k] = topd[r][k];
            out_idx[base + k]  = topi[r][k];
        }
    }
}

// ---------------- phi weights + first denoise -------------------------------
__global__ void phi_denoise1_kernel(const float* __restrict__ normals,
                                    const float* __restrict__ knn_dist,
                                    const int* __restrict__ knn_idx,
                                    float* __restrict__ phi,
                                    float* __restrict__ n1, int P, int N) {
    int i = blockIdx.x * blockDim.x + threadIdx.x;
    if (i >= N) return;
    const int b = i / P;
    const float* nB = normals + (size_t)b * P * 3;
    const size_t base = (size_t)i * K_NN;

    const float s = eps_denom(knn_dist[base] * 2.0f * FILTER_SCALE * FILTER_SCALE);
    float sx = 0.f, sy = 0.f, sz = 0.f, sw = 0.f;
#pragma unroll
    for (int k = 0; k < K_NN; ++k) {
        const float d  = knn_dist[base + k];
        const float w  = fmaxf(1.0f - d / s, 0.0f);
        const float p4 = (w * w) * (w * w);
        phi[base + k] = p4;
        const float* nn = nB + (size_t)knn_idx[base + k] * 3;
        sx += p4 * nn[0]; sy += p4 * nn[1]; sz += p4 * nn[2];
        sw += p4;
    }
    sw = eps_denom(sw);
    n1[(size_t)i * 3 + 0] = sx / sw;
    n1[(size_t)i * 3 + 1] = sy / sw;
    n1[(size_t)i * 3 + 2] = sz / sw;
}

// ---------------- normal_w + second denoise ---------------------------------
__global__ void denoise2_kernel(const float* __restrict__ n1,
                                const float* __restrict__ phi,
                                const int* __restrict__ knn_idx,
                                float* __restrict__ normal_w,
                                float* __restrict__ n2, int P, int N) {
    int i = blockIdx.x * blockDim.x + threadIdx.x;
    if (i >= N) return;
    const int b = i / P;
    const float* n1B = n1 + (size_t)b * P * 3;
    const size_t base = (size_t)i * K_NN;
    const float inv_sigma = 1.0f / (SHARPNESS_SIGMA * SHARPNESS_SIGMA);

    const float px = n1[(size_t)i * 3 + 0];
    const float py = n1[(size_t)i * 3 + 1];
    const float pz = n1[(size_t)i * 3 + 2];
    float pn = sqrtf(px * px + py * py + pz * pz);
    pn = fmaxf(pn, 1e-12f);
    const float ux = px / pn, uy = py / pn, uz = pz / pn;

    float sx = 0.f, sy = 0.f, sz = 0.f, sw = 0.f;
#pragma unroll
    for (int k = 0; k < K_NN; ++k) {
        const float* nk = n1B + (size_t)knn_idx[base + k] * 3;
        const float kx = nk[0], ky = nk[1], kz = nk[2];
        float kn = sqrtf(kx * kx + ky * ky + kz * kz);
        kn = fmaxf(kn, 1e-12f);
        const float dx = kx / kn - ux, dy = ky / kn - uy, dz = kz / kn - uz;
        const float w = __expf(-(dx * dx + dy * dy + dz * dz) * inv_sigma);
        normal_w[base + k] = w;
        const float ww = phi[base + k] * w;
        sx += ww * kx; sy += ww * ky; sz += ww * kz;   // unnormalized n1 neighbors
        sw += ww;
    }
    sw = eps_denom(sw);
    n2[(size_t)i * 3 + 0] = sx / sw;
    n2[(size_t)i * 3 + 1] = sy / sw;
    n2[(size_t)i * 3 + 2] = sz / sw;
}

// ---------------- projection loss per point ---------------------------------
__global__ void loss_kernel(const float* __restrict__ pts,
                            const float* __restrict__ knn_dist,
                            const int* __restrict__ knn_idx,
                            const float* __restrict__ phi,
                            const float* __restrict__ normal_w,
                            const float* __restrict__ n2,
                            float* __restrict__ lossbuf, int P, int N) {
    int i = blockIdx.x * blockDim.x + threadIdx.x;
    if (i >= N) return;
    const int b = i / P;
    const float* ptsB = pts + (size_t)b * P * 3;
    const float* n2B  = n2 + (size_t)b * P * 3;
    const size_t base = (size_t)i * K_NN;

    const float px = pts[(size_t)i * 3 + 0];
    const float py = pts[(size_t)i * 3 + 1];
    const float pz = pts[(size_t)i * 3 + 2];
    const float thr = FILTER_SCALE * knn_dist[base] * 2.0f;   // ball-query radius

    float num = 0.f, den = 0.f;
#pragma unroll
    for (int k = 0; k < K_NN; ++k) {
        float w = phi[base + k] * normal_w[base + k];
        if (knn_dist[base + k] > thr) w = 0.0f;               // ball mask
        const int id = knn_idx[base + k];
        const float* nn = ptsB + (size_t)id * 3;
        const float* nr = n2B + (size_t)id * 3;
        const float dts = (nn[0] - px) * nr[0] + (nn[1] - py) * nr[1] + (nn[2] - pz) * nr[2];
        num += dts * dts * w;
        den += w;
    }
    lossbuf[i] = num / eps_denom(den);
}

// ---------------- deterministic mean reduce ---------------------------------
__global__ __launch_bounds__(256)
void reduce_kernel(const float* __restrict__ lossbuf, float* __restrict__ out, int N) {
    __shared__ float sm[256];
    float acc = 0.f;
    for (int i = threadIdx.x; i < N; i += 256) acc += lossbuf[i];
    sm[threadIdx.x] = acc;
    __syncthreads();
    for (int s = 128; s > 0; s >>= 1) {
        if ((int)threadIdx.x < s) sm[threadIdx.x] += sm[threadIdx.x + s];
        __syncthreads();
    }
    if (threadIdx.x == 0) out[0] = sm[0] / (float)N;
}

// ---------------------------------------------------------------------------
extern "C" void kernel_launch(void* const* d_in, const int* in_sizes, int n_in,
                              void* d_out, int out_size, void* d_ws, size_t ws_size,
                              hipStream_t stream) {
    const float* points  = (const float*)d_in[0];
    const float* normals = (const float*)d_in[1];
    float* out = (float*)d_out;

    const int P = 8192;
    const int N = in_sizes[0] / 3;   // B*P
    const int B = N / P;

    char* ws = (char*)d_ws;
    float* knn_dist = (float*)ws; ws += (size_t)N * K_NN * sizeof(float);
    int*   knn_idx  = (int*)ws;   ws += (size_t)N * K_NN * sizeof(int);
    float* phi      = (float*)ws; ws += (size_t)N * K_NN * sizeof(float);
    float* n1       = (float*)ws; ws += (size_t)N * 3 * sizeof(float);
    float* nw       = (float*)ws; ws += (size_t)N * K_NN * sizeof(float);
    float* n2       = (float*)ws; ws += (size_t)N * 3 * sizeof(float);
    float* lossbuf  = (float*)ws; ws += (size_t)N * sizeof(float);

    const int T = 256;

    dim3 kg(P / 16, B);
    knn_kernel<<<kg, 32, 0, stream>>>(points, knn_dist, knn_idx, P);

    phi_denoise1_kernel<<<(N + T - 1) / T, T, 0, stream>>>(normals, knn_dist, knn_idx,
                                                           phi, n1, P, N);
    denoise2_kernel<<<(N + T - 1) / T, T, 0, stream>>>(n1, phi, knn_idx, nw, n2, P, N);
    loss_kernel<<<(N + T - 1) / T, T, 0, stream>>>(points, knn_dist, knn_idx, phi, nw,
                                                   n2, lossbuf, P, N);
    reduce_kernel<<<1, 256, 0, stream>>>(lossbuf, out, N);
}